// DDI_decagon_prob_84482006713340
// MI455X (gfx1250) — compile-verified
//
#include <hip/hip_runtime.h>
#include <cstddef>

typedef float v2f __attribute__((ext_vector_type(2)));
typedef float v8f __attribute__((ext_vector_type(8)));

#define RGCN_R   128
#define RGCN_B   4
#define RGCN_IN  128
#define RGCN_HID 64
#define RGCN_W   (RGCN_HID * (RGCN_B + 1))   // 320 output cols per layer GEMM

__device__ __forceinline__ v8f wmma_f32x4(v2f a, v2f b, v8f c) {
  // V_WMMA_F32_16X16X4_F32 : D(16x16 f32) = A(16x4 f32) * B(4x16 f32) + C
  return __builtin_amdgcn_wmma_f32_16x16x4_f32(false, a, false, b, (short)0, c,
                                               false, false);
}

// ---------------------------------------------------------------------------
// Per-(dst,relation) edge count:  cnt[dst*R + etype] += 1
// ---------------------------------------------------------------------------
__global__ void rgcn_count_kernel(const int* __restrict__ dst,
                                  const int* __restrict__ et,
                                  float* __restrict__ cnt, int E) {
  int e = blockIdx.x * blockDim.x + threadIdx.x;
  if (e < E) atomicAdd(&cnt[(size_t)dst[e] * RGCN_R + et[e]], 1.0f);
}

// ---------------------------------------------------------------------------
// Pack [root | basis0..3] (K x 320) into WMMA-fragment layout:
//   Bp[(k>>2)*1280 + n*4 + (k&3)] = B[k][n]
// so each lane's two K-values for a fragment are one b64 load.
// ---------------------------------------------------------------------------
template<int K>
__global__ void rgcn_pack_kernel(const float* __restrict__ rootW,
                                 const float* __restrict__ basisW,
                                 float* __restrict__ Bp) {
  const int t = blockIdx.x * blockDim.x + threadIdx.x;
  if (t >= K * RGCN_W) return;
  const int m = t & 3;
  const int n = (t >> 2) % RGCN_W;
  const int q = t / (4 * RGCN_W);
  const int k = q * 4 + m;
  float v;
  if (n < RGCN_HID) {
    v = rootW[k * RGCN_HID + n];
  } else {
    const int b = (n - RGCN_HID) >> 6;
    const int c = (n - RGCN_HID) & 63;
    v = basisW[((size_t)b * K + k) * RGCN_HID + c];
  }
  Bp[t] = v;
}

// ---------------------------------------------------------------------------
// Fused GEMM:  A[N x K]  x  packed [root | basis0..3]  (K x 320)
//   tileN 0..3  -> h[n, col] = A@root + bias     (self/root transform)
//   tileN 4..19 -> z[n, col-64] = A@basis_b      (basis projections)
// One wave per 16x16 tile; f32 WMMA 16x16x4; A and B fragments are b64 loads.
// A-frag: lanes 0-15 hold K={k,k+1}, lanes 16-31 hold K={k+2,k+3}.
// C/D:    VGPR v = rows v (lanes 0-15) / v+8 (lanes 16-31).
// ---------------------------------------------------------------------------
template<int K, bool RELU_A>
__global__ void rgcn_gemm_kernel(const float* __restrict__ A,
                                 const float* __restrict__ Bp,
                                 const float* __restrict__ bias,
                                 float* __restrict__ h,
                                 float* __restrict__ z,
                                 int nTilesM) {
  const int wave = threadIdx.x >> 5;
  const int lane = threadIdx.x & 31;
  const int hi   = lane >> 4;          // 0: lanes 0-15, 1: lanes 16-31
  const int lm   = lane & 15;
  const int tileM = blockIdx.x * (blockDim.x >> 5) + wave;
  const int tileN = blockIdx.y;        // 0..19
  if (tileM >= nTilesM) return;

  const int row = tileM * 16 + lm;
  const float* __restrict__ arow = A + (size_t)row * K + 2 * hi;
  const int n = tileN * 16 + lm;
  const float* __restrict__ bptr = Bp + (size_t)n * 4 + 2 * hi;

  v8f acc = {};
#pragma unroll 8
  for (int k = 0; k < K; k += 4) {
    v2f a = *(const v2f*)(arow + k);
    if (RELU_A) { a.x = fmaxf(a.x, 0.f); a.y = fmaxf(a.y, 0.f); }
    v2f b = *(const v2f*)(bptr + (size_t)(k >> 2) * (4 * RGCN_W));
    acc = wmma_f32x4(a, b, acc);
  }

  if (tileN < 4) {
    const float bv = bias[n];
#pragma unroll
    for (int v = 0; v < 8; ++v) {
      const int r = tileM * 16 + v + 8 * hi;
      h[(size_t)r * RGCN_HID + n] = acc[v] + bv;
    }
  } else {
    const int zc = n - RGCN_HID;
#pragma unroll
    for (int v = 0; v < 8; ++v) {
      const int r = tileM * 16 + v + 8 * hi;
      z[(size_t)r * (RGCN_B * RGCN_HID) + zc] = acc[v];
    }
  }
}

// ---------------------------------------------------------------------------
// Edge scatter: one wave per edge.
//   msg[h] = sum_b comp[etype,b] * norm * gate * z[src, b*64+h]
//   atomicAdd(hout[dst, h], msg[h])   (h split across 32 lanes, 2 each)
// ---------------------------------------------------------------------------
__global__ void rgcn_scatter_kernel(const int* __restrict__ src,
                                    const int* __restrict__ dst,
                                    const int* __restrict__ et,
                                    const float* __restrict__ comp,
                                    const float* __restrict__ probs,
                                    const float* __restrict__ cnt,
                                    const float* __restrict__ z,
                                    float* __restrict__ hout, int E) {
  const int e = blockIdx.x * (blockDim.x >> 5) + (threadIdx.x >> 5);
  if (e >= E) return;
  const int lane = threadIdx.x & 31;

  const int s = src[e], d = dst[e], t = et[e];
  const float gate = 1.f / (1.f + __expf(-probs[t]));
  const float c    = cnt[(size_t)d * RGCN_R + t];
  const float g    = gate / fmaxf(c, 1.f);
  const float c0 = comp[t * RGCN_B + 0] * g;
  const float c1 = comp[t * RGCN_B + 1] * g;
  const float c2 = comp[t * RGCN_B + 2] * g;
  const float c3 = comp[t * RGCN_B + 3] * g;

  const float* __restrict__ zp = z + (size_t)s * (RGCN_B * RGCN_HID);
  float* __restrict__ hp = hout + (size_t)d * RGCN_HID;
#pragma unroll
  for (int h0 = lane; h0 < RGCN_HID; h0 += 32) {
    const float m = c0 * zp[h0] + c1 * zp[RGCN_HID + h0] +
                    c2 * zp[2 * RGCN_HID + h0] + c3 * zp[3 * RGCN_HID + h0];
    atomicAdd(&hp[h0], m);
  }
}

// ---------------------------------------------------------------------------
// Scorer weights, fragment-packed:
//   Wtp[(k>>2)*512 + r*4 + (k&3)] = D[r,i]*Rm[i,j]*D[r,j],  k = i*64+j
// Thread mapping chosen so writes are fully coalesced.
// ---------------------------------------------------------------------------
__global__ void rgcn_wmat_kernel(const float* __restrict__ Rm,
                                 const float* __restrict__ D,
                                 float* __restrict__ Wtp) {
  const int t = blockIdx.x * blockDim.x + threadIdx.x;
  if (t >= RGCN_HID * RGCN_HID * RGCN_R) return;
  const int m = t & 3;
  const int r = (t >> 2) & (RGCN_R - 1);
  const int q = t >> 9;
  const int k = q * 4 + m;
  const int i = k >> 6, j = k & 63;
  Wtp[t] = D[r * RGCN_HID + i] * Rm[i * RGCN_HID + j] * D[r * RGCN_HID + j];
}

// ---------------------------------------------------------------------------
// Scorer: out[e,r] = sigmoid( sum_k (x1[e,i(k)]*x2[e,j(k)]) * W[k,r] )
// GEMM M=ET, N=128, K=4096; A generated on the fly from LDS-staged features.
// Block = 256 threads = 8 waves; block owns 32 target edges (two 16-row
// M-tiles -> two accumulators sharing each B fragment); wave w owns relation
// tile w. B fragments are single b64 loads from the packed Wtp (L2-resident).
// LDS padded [32][65] to kill bank conflicts.
// ---------------------------------------------------------------------------
__global__ void rgcn_score_kernel(const float* __restrict__ hB,
                                  const int* __restrict__ tei, // [2*ET]
                                  const float* __restrict__ Wtp,
                                  float* __restrict__ out, int ET) {
  __shared__ float x1s[32][65];
  __shared__ float x2s[32][65];
  const int base = blockIdx.x * 32;

  for (int idx = threadIdx.x; idx < 32 * RGCN_HID; idx += blockDim.x) {
    const int m = idx >> 6, i = idx & 63;
    int e = base + m;
    if (e >= ET) e = 0;                    // clamp; stores are guarded below
    const int n1 = tei[e], n2 = tei[ET + e];
    x1s[m][i] = fmaxf(hB[(size_t)n1 * RGCN_HID + i], 0.f);  // fused ReLU
    x2s[m][i] = fmaxf(hB[(size_t)n2 * RGCN_HID + i], 0.f);
  }
  __syncthreads();

  const int wave = threadIdx.x >> 5;       // relation tile 0..7
  const int lane = threadIdx.x & 31;
  const int hi   = lane >> 4;
  const int lm   = lane & 15;
  const int r    = wave * 16 + lm;
  const float* __restrict__ wbase = Wtp + (size_t)r * 4 + 2 * hi;

  v8f acc0 = {}, acc1 = {};
  for (int i = 0; i < RGCN_HID; ++i) {
    const float xi0 = x1s[lm][i];
    const float xi1 = x1s[16 + lm][i];
    const float* __restrict__ wrow = wbase + (size_t)i * (16 * 512);
#pragma unroll
    for (int jj = 0; jj < RGCN_HID; jj += 4) {
      const int j0 = jj + 2 * hi;          // stays inside this i-block
      v2f b = *(const v2f*)(wrow + (size_t)(jj >> 2) * 512);
      v2f a0, a1;
      a0.x = xi0 * x2s[lm][j0];      a0.y = xi0 * x2s[lm][j0 + 1];
      a1.x = xi1 * x2s[16 + lm][j0]; a1.y = xi1 * x2s[16 + lm][j0 + 1];
      acc0 = wmma_f32x4(a0, b, acc0);
      acc1 = wmma_f32x4(a1, b, acc1);
    }
  }

#pragma unroll
  for (int v = 0; v < 8; ++v) {
    const int e0 = base + v + 8 * hi;
    const int e1 = base + 16 + v + 8 * hi;
    if (e0 < ET) out[(size_t)e0 * RGCN_R + r] = 1.f / (1.f + __expf(-acc0[v]));
    if (e1 < ET) out[(size_t)e1 * RGCN_R + r] = 1.f / (1.f + __expf(-acc1[v]));
  }
}

// ---------------------------------------------------------------------------
extern "C" void kernel_launch(void* const* d_in, const int* in_sizes, int n_in,
                              void* d_out, int out_size, void* d_ws,
                              size_t ws_size, hipStream_t stream) {
  const float* x      = (const float*)d_in[0];
  const int*   eidx   = (const int*)d_in[1];
  const int*   etype  = (const int*)d_in[2];
  const int*   tei    = (const int*)d_in[3];
  const float* probs  = (const float*)d_in[4];
  const float* basis1 = (const float*)d_in[5];
  const float* comp1  = (const float*)d_in[6];
  const float* root1  = (const float*)d_in[7];
  const float* bias1  = (const float*)d_in[8];
  const float* basis2 = (const float*)d_in[9];
  const float* comp2  = (const float*)d_in[10];
  const float* root2  = (const float*)d_in[11];
  const float* bias2  = (const float*)d_in[12];
  const float* Rm     = (const float*)d_in[13];
  const float* D      = (const float*)d_in[14];

  const int N  = in_sizes[0] / RGCN_IN;    // 50000
  const int E  = in_sizes[2];              // 1,000,000
  const int ET = in_sizes[3] / 2;          // 10,000

  // Workspace (floats):
  //   cnt[N*128] | z[N*256] | hA[N*64] | hB[N*64] | Wtp[4096*128]
  //   | Bp1[128*320] | Bp2[64*320]
  float* cnt = (float*)d_ws;
  float* z   = cnt + (size_t)N * RGCN_R;
  float* hA  = z   + (size_t)N * RGCN_B * RGCN_HID;
  float* hB  = hA  + (size_t)N * RGCN_HID;
  float* Wtp = hB  + (size_t)N * RGCN_HID;
  float* Bp1 = Wtp + (size_t)RGCN_HID * RGCN_HID * RGCN_R;
  float* Bp2 = Bp1 + (size_t)RGCN_IN * RGCN_W;

  const int* src = eidx;
  const int* dst = eidx + E;

  hipMemsetAsync(cnt, 0, (size_t)N * RGCN_R * sizeof(float), stream);

  rgcn_count_kernel<<<(E + 255) / 256, 256, 0, stream>>>(dst, etype, cnt, E);

  rgcn_pack_kernel<RGCN_IN>
      <<<(RGCN_IN * RGCN_W + 255) / 256, 256, 0, stream>>>(root1, basis1, Bp1);
  rgcn_pack_kernel<RGCN_HID>
      <<<(RGCN_HID * RGCN_W + 255) / 256, 256, 0, stream>>>(root2, basis2, Bp2);

  const int tilesM = N / 16;               // 3125 (N divisible by 16)
  dim3 gg((tilesM + 3) / 4, RGCN_W / 16);  // 4 waves/block x 20 N-tiles

  // Layer 1: hA = x@root1 + bias1 ; z = x@basis1[b]
  rgcn_gemm_kernel<RGCN_IN, false>
      <<<gg, 128, 0, stream>>>(x, Bp1, bias1, hA, z, tilesM);
  rgcn_scatter_kernel<<<(E + 7) / 8, 256, 0, stream>>>(src, dst, etype, comp1,
                                                       probs, cnt, z, hA, E);

  // Layer 2: A = relu(hA); hB = A@root2 + bias2 ; z = A@basis2[b]
  rgcn_gemm_kernel<RGCN_HID, true>
      <<<gg, 128, 0, stream>>>(hA, Bp2, bias2, hB, z, tilesM);
  rgcn_scatter_kernel<<<(E + 7) / 8, 256, 0, stream>>>(src, dst, etype, comp2,
                                                       probs, cnt, z, hB, E);

  // Scorer weights (packed) + bilinear scorer (ReLU fused into gather)
  rgcn_wmat_kernel<<<(RGCN_HID * RGCN_HID * RGCN_R + 255) / 256, 256, 0,
                     stream>>>(Rm, D, Wtp);
  rgcn_score_kernel<<<(ET + 31) / 32, 256, 0, stream>>>(hB, tei, Wtp,
                                                        (float*)d_out, ET);
}